// barrier_controller_13907104104456
// MI455X (gfx1250) — compile-verified
//
#include <hip/hip_runtime.h>
#include <hip/hip_bf16.h>

typedef __attribute__((ext_vector_type(2))) float v2f;
typedef __attribute__((ext_vector_type(8))) float v8f;

#define ALPHA1 1.2566f
#define ALPHA2 1.5f
#define ALPHA3 0.9f
#define S_STAR 20.0f
#define TAU    4.0f

// ------------------------------------------------------------------
// Kernel 1: collapse the (linear!) MLP  W3@W2@W1 -> Weff (rows 0..3),
// W3@(W2@b1+b2)+b3 -> beff.  ws[0..23] = Weff[4x6], ws[24..27] = beff.
// ------------------------------------------------------------------
__global__ void collapse_mlp_kernel(const float* __restrict__ W1, const float* __restrict__ b1,
                                    const float* __restrict__ W2, const float* __restrict__ b2,
                                    const float* __restrict__ W3, const float* __restrict__ b3,
                                    float* __restrict__ ws) {
  __shared__ float T[100][6];   // W2 @ W1  (100 x 6)
  __shared__ float tb[100];     // W2 @ b1 + b2
  int j = threadIdx.x;
  if (j < 100) {
    float acc[6] = {0.f, 0.f, 0.f, 0.f, 0.f, 0.f};
    float accb = 0.f;
    for (int m = 0; m < 50; ++m) {
      float w = W2[j * 50 + m];
      accb += w * b1[m];
#pragma unroll
      for (int k = 0; k < 6; ++k) acc[k] += w * W1[m * 6 + k];
    }
#pragma unroll
    for (int k = 0; k < 6; ++k) T[j][k] = acc[k];
    tb[j] = accb + b2[j];
  }
  __syncthreads();
  if (j < 24) {                       // Weff rows 0..3 (only u0..u3 are used)
    int r = j / 6, k = j % 6;
    float acc = 0.f;
    for (int q = 0; q < 100; ++q) acc += W3[r * 100 + q] * T[q][k];
    ws[j] = acc;
  }
  if (j >= 32 && j < 36) {            // beff rows 0..3
    int r = j - 32;
    float acc = 0.f;
    for (int q = 0; q < 100; ++q) acc += W3[r * 100 + q] * tb[q];
    ws[24 + r] = acc + b3[r];
  }
}

// ------------------------------------------------------------------
// Kernel 2: fused pass. Per wave: one 16-row tile.
//   u(16 rows x 4 neurons) via two chained V_WMMA_F32_16X16X4_F32
//   (K = 6 split as 4 + 2).  D layout: lane n (<16) holds row base+n,
//   c[0..3] = u0..u3 for that row -> scalar epilogue entirely in-lane.
//   B-operand: divergence-free b64 loads (per-lane address), mask by
//   select, so the inner loop has no exec-mask branching.
// ------------------------------------------------------------------
__global__ void fused_controller_kernel(const float* __restrict__ zf,
                                        const float* __restrict__ ws,
                                        float* __restrict__ out, int n) {
  const int  lane = threadIdx.x & 31;
  const int  t    = lane & 15;
  const bool hi   = lane >= 16;
  const int  wavesPerBlock = blockDim.x >> 5;
  const int  gwave    = blockIdx.x * wavesPerBlock + (threadIdx.x >> 5);
  const int  numWaves = gridDim.x * wavesPerBlock;
  const int  tiles    = (n + 15) >> 4;

  // ---- A operand: Weff padded to 16 rows; K-chunks {0..3} and {4,5,0,0}.
  // A 16x4 layout: lanes 0-15 -> (K0,K1), lanes 16-31 -> (K2,K3), M = lane&15.
  float w0 = 0.f, w1 = 0.f, w2 = 0.f, w3 = 0.f, w4 = 0.f, w5 = 0.f;
  if (t < 4) {
    w0 = ws[t * 6 + 0]; w1 = ws[t * 6 + 1]; w2 = ws[t * 6 + 2];
    w3 = ws[t * 6 + 3]; w4 = ws[t * 6 + 4]; w5 = ws[t * 6 + 5];
  }
  v2f aK0, aK1;
  aK0.x = hi ? w2 : w0;   aK0.y = hi ? w3 : w1;
  aK1.x = hi ? 0.f : w4;  aK1.y = hi ? 0.f : w5;

  const float be0 = ws[24], be1 = ws[25], be2 = ws[26], be3 = ws[27];

  // ---- compile-time folded row-vectors (d2@f, d3@f, (d3@f)@f)
  const float c1 = 1.f - TAU * ALPHA3;
  const float c2 = -TAU * ALPHA1;
  const float c3 = -1.f + TAU * ALPHA2;
  const float e1 = TAU * ALPHA3;
  const float e2 = TAU * ALPHA1;
  const float e3 = 1.f - TAU * ALPHA2 - TAU * ALPHA3;
  const float e4 = -TAU * ALPHA1;
  const float e5 = -1.f + TAU * ALPHA2;
  const float g1 = e2 + e3 * ALPHA3;
  const float g2 = e3 * ALPHA1;
  const float g3 = -e2 - e3 * ALPHA2 + e4 + e5 * ALPHA3;
  const float g4 = e5 * ALPHA1;
  const float g5 = -e4 - e5 * ALPHA2;
  const float invA0 = -1.f / TAU;            // A = [-tau, -tau*a3]: both < 0
  const float invA1 = -1.f / (TAU * ALPHA3); // -> ub = +1e30, lb = max(h/A)

  const size_t N = (size_t)n;
  const float* __restrict__ x1p = zf + N;      // X = z.reshape(6, B) rows 1..5
  const float* __restrict__ x2p = zf + 2 * N;
  const float* __restrict__ x3p = zf + 3 * N;
  const float* __restrict__ x4p = zf + 4 * N;
  const float* __restrict__ x5p = zf + 5 * N;

  const int koff = hi ? 2 : 0;  // per-lane K-pair offset into the z row

  for (int tile = gwave; tile < tiles; tile += numWaves) {
    int row = tile * 16 + t;
    int rc  = row < n ? row : n - 1;
    const float* zr = zf + (size_t)rc * 6;

    // {z0,z1} on lanes 0-15, {z2,z3} on lanes 16-31: one b64, no branches.
    v2f bK0 = *reinterpret_cast<const v2f*>(zr + koff);
    // {z4,z5} loaded by all lanes (same cachelines), zeroed on high lanes.
    v2f tail = *reinterpret_cast<const v2f*>(zr + 4);
    v2f bK1;
    bK1.x = hi ? 0.f : tail.x;
    bK1.y = hi ? 0.f : tail.y;

    v8f c = {0.f, 0.f, 0.f, 0.f, 0.f, 0.f, 0.f, 0.f};
    c = __builtin_amdgcn_wmma_f32_16x16x4_f32(false, aK0, false, bK0,
                                              (short)0, c, false, false);
    c = __builtin_amdgcn_wmma_f32_16x16x4_f32(false, aK1, false, bK1,
                                              (short)0, c, false, false);

    if (!hi && row < n) {
      float u0 = c[0] + be0;
      float p1 = c[1] + be1;
      float p2 = c[2] + be2;
      float p3 = c[3] + be3;
      // X column `row` of z.reshape(6, B): strided, fully coalesced.
      float X1 = x1p[row];
      float X2 = x2p[row];
      float X3 = x3p[row];
      float X4 = x4p[row];
      float X5 = x5p[row];

      float eta1  = X2 + S_STAR - TAU * (X3 - X1);
      float bb1   = c1 * X1 + c2 * X2 + c3 * X3 + p1 * eta1;
      float eta2a = X4 + S_STAR - TAU * (X5 - X3);
      float eta2b = e1 * X1 + e2 * X2 + e3 * X3 + e4 * X4 + e5 * X5;
      float bb2   = g1 * X1 + g2 * X2 + g3 * X3 + g4 * X4 + g5 * X5
                  + p2 * eta2a + p3 * eta2b;

      float lb  = fmaxf(bb1 * invA0, bb2 * invA1);
      float res = fminf(fmaxf(2.f * u0, lb), 1e30f);
      out[row] = res;
    }
  }
}

extern "C" void kernel_launch(void* const* d_in, const int* in_sizes, int n_in,
                              void* d_out, int out_size, void* d_ws, size_t ws_size,
                              hipStream_t stream) {
  const float* z  = (const float*)d_in[0];
  // d_in[1] = tilde_vh (unused by the reference math)
  const float* W1 = (const float*)d_in[2];
  const float* b1 = (const float*)d_in[3];
  const float* W2 = (const float*)d_in[4];
  const float* b2 = (const float*)d_in[5];
  const float* W3 = (const float*)d_in[6];
  const float* b3 = (const float*)d_in[7];
  float* out = (float*)d_out;
  float* ws  = (float*)d_ws;
  int n = out_size;  // B rows, output is (B,1) float32

  collapse_mlp_kernel<<<1, 128, 0, stream>>>(W1, b1, W2, b2, W3, b3, ws);

  int tiles  = (n + 15) / 16;
  int blocks = (tiles + 7) / 8;          // 8 waves / 256-thread block
  if (blocks > 4096) blocks = 4096;      // grid-stride covers the rest
  if (blocks < 1) blocks = 1;
  fused_controller_kernel<<<blocks, 256, 0, stream>>>(z, ws, out, n);
}